// fixedDilatedConv_73658689126476
// MI455X (gfx1250) — compile-verified
//
#include <hip/hip_runtime.h>

// ---------------------------------------------------------------------------
// fixedDilatedConv on MI455X (gfx1250)
//
// out[b,k,t] = w1*x[b,c1,t+p1] + w2*x[b,c2,t+p2]   (exactly 2 taps per filter)
//
// Store-bandwidth bound: 626 MB of fp32 output @ 23.3 TB/s ~= 27 us floor.
// Design: async-stage x tile (+halo) into LDS once per block (gfx1250
// GLOBAL_LOAD_ASYNC_TO_LDS_B128 + s_wait_asynccnt), then stream aligned
// b128 non-temporal stores for 120 filters x 1024 positions per block.
// NOTE: conv_sparse is defined FIRST so the disasm snippet shows the hot loop.
// ---------------------------------------------------------------------------

#define NFILT   600
#define LCONV   8192
#define LOUT    8153          // 8192 - 39
#define KSIZE   40
#define TILE    1024
#define HALO    39
#define SMSTRIDE 1064         // TILE + HALO + 1 ; multiple of 4 -> 16B aligned rows
#define KSPLIT  5
#define KCHUNK  (NFILT / KSPLIT)   // 120
#define NTILES  8                  // ceil(8153 / 1024)

struct TapMeta { int off1; int off2; float w1; float w2; };

// ---- detect gfx1250 async global->LDS builtins -----------------------------
#ifndef USE_ASYNC_LDS
#if defined(__has_builtin)
#  if __has_builtin(__builtin_amdgcn_global_load_async_to_lds_b128) && \
      __has_builtin(__builtin_amdgcn_s_wait_asynccnt)
#    define USE_ASYNC_LDS 1
#  else
#    define USE_ASYNC_LDS 0
#  endif
#else
#  define USE_ASYNC_LDS 0
#endif
#endif

// Builtin signature (from clang diagnostic): param 0 is
//   'int __vector(4) __device__ *'  ==  v4i in addrspace(1)
typedef int   v4i __attribute__((vector_size(16)));
typedef float v4f __attribute__((ext_vector_type(4)));
typedef __attribute__((address_space(1))) v4i* gv4i_ptr;   // global int4*
typedef __attribute__((address_space(3))) v4i* lv4i_ptr;   // lds int4*

// ---------------------------------------------------------------------------
// Main kernel (hot loop).
//   grid = (NTILES=8, B=32, KSPLIT=5), block = 256 threads (8 waves).
//   Each block: async-stage x[b, 0..3, t0 .. t0+1062] into LDS, then for
//   k in [k0, k0+120): out row = w1*sm[off1+t] + w2*sm[off2+t].
//   Stores: per-row alignment fixup (row stride 8153 is odd), then one
//   16B-aligned b128 NT store per lane -> 512B contiguous per wave.
//   Tile covers global positions [t0, t0+1024); the aligned vector region
//   overlaps the next tile's head, which that tile writes itself, so we
//   clip at t < TILE -> every element written exactly once.
// ---------------------------------------------------------------------------
__global__ __launch_bounds__(256) void conv_sparse(
    const float* __restrict__ x,
    const TapMeta* __restrict__ meta,
    float* __restrict__ out) {
  __shared__ float sm[4 * SMSTRIDE];

  const int tid = threadIdx.x;
  const int t0  = blockIdx.x * TILE;
  const int b   = blockIdx.y;
  const int k0  = blockIdx.z * KCHUNK;

  const float* xb = x + (size_t)b * 4 * LCONV;

  // ---- stage x tile (+halo) into LDS --------------------------------------
  for (int c = 0; c < 4; ++c) {
    const float* g = xb + c * LCONV + t0;
    for (int j = tid * 4; j < SMSTRIDE; j += 256 * 4) {
#if USE_ASYNC_LDS
      if (t0 + j + 3 < LCONV) {
        // gfx1250 async DMA: 16B global -> LDS, tracked by ASYNCcnt
        __builtin_amdgcn_global_load_async_to_lds_b128(
            (gv4i_ptr)(g + j),
            (lv4i_ptr)&sm[c * SMSTRIDE + j],
            /*offset=*/0, /*cpol=*/0);
      } else {
        #pragma unroll
        for (int e = 0; e < 4; ++e)
          sm[c * SMSTRIDE + j + e] = (t0 + j + e < LCONV) ? g[j + e] : 0.f;
      }
#else
      #pragma unroll
      for (int e = 0; e < 4; ++e)
        sm[c * SMSTRIDE + j + e] = (t0 + j + e < LCONV) ? g[j + e] : 0.f;
#endif
    }
  }
#if USE_ASYNC_LDS
  __builtin_amdgcn_s_wait_asynccnt(0);   // my async writes to LDS landed
#endif
  __syncthreads();                       // everyone's writes visible

  const int lim = LOUT - t0;             // valid positions in this tile

  // ---- stream 120 filters x 1024 positions --------------------------------
  for (int k = k0; k < k0 + KCHUNK; ++k) {
    const TapMeta m = meta[k];           // uniform across block -> scalar loads
    const float* s1 = sm + m.off1;
    const float* s2 = sm + m.off2;
    const size_t base = ((size_t)b * NFILT + k) * LOUT + (size_t)t0;
    float* o = out + base;
    const int a = (int)((4u - ((unsigned)base & 3u)) & 3u);  // head fixup (SALU)

    // head scalars: global positions [t0, t0+a)
    if (tid < a && tid < lim) {
      __builtin_nontemporal_store(m.w1 * s1[tid] + m.w2 * s2[tid], o + tid);
    }

    const int tv = a + tid * 4;          // 16B-aligned chunk start
    if (tv + 3 < TILE && tv + 3 < lim) {
      v4f r;
      r.x = m.w1 * s1[tv + 0] + m.w2 * s2[tv + 0];
      r.y = m.w1 * s1[tv + 1] + m.w2 * s2[tv + 1];
      r.z = m.w1 * s1[tv + 2] + m.w2 * s2[tv + 2];
      r.w = m.w1 * s1[tv + 3] + m.w2 * s2[tv + 3];
      __builtin_nontemporal_store(r, (v4f*)(o + tv));   // global_store_b128 nt
    } else {
      #pragma unroll
      for (int e = 0; e < 4; ++e) {
        const int t = tv + e;
        if (t < TILE && t < lim)
          __builtin_nontemporal_store(m.w1 * s1[t] + m.w2 * s2[t], o + t);
      }
    }
  }
}

// ---------------------------------------------------------------------------
// Prepass: extract the 2 nonzero taps of each filter from the dense [600,4,40]
// weight tensor.  Offsets are pre-baked into the LDS tile layout
// (c * SMSTRIDE + p) so the hot loop does zero address math beyond an add.
// ---------------------------------------------------------------------------
__global__ void extract_taps(const float* __restrict__ W,
                             TapMeta* __restrict__ meta) {
  int k = blockIdx.x * blockDim.x + threadIdx.x;
  if (k >= NFILT) return;
  int   off[2] = {0, 0};
  float wv[2]  = {0.f, 0.f};
  int   n = 0;
  for (int c = 0; c < 4; ++c) {
    for (int p = 0; p < KSIZE; ++p) {
      float v = W[(k * 4 + c) * KSIZE + p];
      if (v != 0.f && n < 2) {
        off[n] = c * SMSTRIDE + p;
        wv[n]  = v;
        ++n;
      }
    }
  }
  if (n == 1) { off[1] = off[0]; wv[1] = 0.f; }  // defensive (shouldn't happen)
  TapMeta m;
  m.off1 = off[0]; m.off2 = off[1]; m.w1 = wv[0]; m.w2 = wv[1];
  meta[k] = m;
}

// ---------------------------------------------------------------------------
extern "C" void kernel_launch(void* const* d_in, const int* in_sizes, int n_in,
                              void* d_out, int out_size, void* d_ws, size_t ws_size,
                              hipStream_t stream) {
  const float* x = (const float*)d_in[0];   // [32, 4, 8192] fp32
  const float* W = (const float*)d_in[1];   // [600, 4, 40] fp32
  float* out = (float*)d_out;               // [32, 600, 8153] fp32
  TapMeta* meta = (TapMeta*)d_ws;           // 600 * 16 B = 9.6 KB scratch

  extract_taps<<<(NFILT + 255) / 256, 256, 0, stream>>>(W, meta);

  dim3 grid(NTILES, 32, KSPLIT);
  conv_sparse<<<grid, 256, 0, stream>>>(x, meta, out);
}